// DNN_WPE_19920058318882
// MI455X (gfx1250) — compile-verified
//
#include <hip/hip_runtime.h>

// ---------------- problem constants (match reference) ----------------
#define Bsz   8
#define Tn    600
#define Cn    8
#define Fn    257
#define NTOT  (Bsz * Fn)        // 2056 independent (b,f) problems
#define TAPS  8
#define DELAYN 3
#define KD    (TAPS * Cn)       // 64 filter coefficients
#define NCOLS 72                // 64 R columns + 8 RHS columns
#define TFRM  590               // Tf = T - DELAY - TAPS + 1
#define KPAD  592               // K padded to multiple of 4 (zero weights)
#define LSTR  608               // LDS row stride for Y (>= 602, zero padded)
#define EPSF  1e-10f
#define PMINF 1e-6f

typedef float v2f __attribute__((ext_vector_type(2)));
typedef float v8f __attribute__((ext_vector_type(8)));

// f32 WMMA: D(16x16) = A(16x4) * B(4x16) + C   -- CDNA5 V_WMMA_F32_16X16X4_F32
static __device__ __forceinline__ v8f wmma4(v2f a, v2f b, v8f c) {
  return __builtin_amdgcn_wmma_f32_16x16x4_f32(false, a, false, b, (short)0, c,
                                               false, false);
}

// ---------------- K_A: (B,T,C,F) -> Y (N=B*F, C, T), planar re/im ----------
__global__ void k_pack(const float* __restrict__ dre,
                       const float* __restrict__ dimg,
                       float* __restrict__ yre, float* __restrict__ yim) {
  __shared__ float tr[16][17], ti[16][17];
  int b = blockIdx.z, c = blockIdx.y;
  int tt = blockIdx.x / 17, ft = blockIdx.x % 17;
  int tx = threadIdx.x, ty = threadIdx.y;
  int t = tt * 16 + ty, f = ft * 16 + tx;           // read: f contiguous
  if (t < Tn && f < Fn) {
    size_t s = ((size_t)(b * Tn + t) * Cn + c) * Fn + f;
    tr[ty][tx] = dre[s];
    ti[ty][tx] = dimg[s];
  }
  __syncthreads();
  int t2 = tt * 16 + tx, f2 = ft * 16 + ty;         // write: t contiguous
  if (t2 < Tn && f2 < Fn) {
    int n = b * Fn + f2;
    size_t o = ((size_t)n * Cn + c) * Tn + t2;
    yre[o] = tr[tx][ty];
    yim[o] = ti[tx][ty];
  }
}

// ---------------- K_B: inverse averaged power -------------------------------
__global__ void k_power(const float* __restrict__ sre,
                        const float* __restrict__ sim,
                        float* __restrict__ invp) {
  int idx = blockIdx.x * blockDim.x + threadIdx.x;
  if (idx >= NTOT * Tn) return;
  int n = idx / Tn, t = idx % Tn;
  const float* pr = sre + (size_t)n * Cn * Tn + t;
  const float* pi = sim + (size_t)n * Cn * Tn + t;
  float s = 0.f;
#pragma unroll
  for (int c = 0; c < Cn; ++c) {
    float a = pr[c * Tn], b = pi[c * Tn];
    s += a * a + b * b;
  }
  float p = s * (1.0f / Cn);
  p = (p < PMINF) ? PMINF : p;
  invp[idx] = 1.0f / p;
}

// ---------------- K_C: R = Psi^H diag(w) [Psi | Ytgt]  (WMMA f32) -----------
// One workgroup per n. R is Hermitian: compute only tiles with ct >= jt
// (10 upper-triangle R tiles + 4 RHS tiles = 14 waves). Complex product via
// 4 real WMMAs per K-chunk; K padded to 592 with zero weights.
#define CORR_T 448              // 14 waves
__global__ void __launch_bounds__(CORR_T) k_corr(
    const float* __restrict__ yre, const float* __restrict__ yim,
    const float* __restrict__ invp,
    float* __restrict__ Rre, float* __restrict__ Rim,
    float* __restrict__ RCre, float* __restrict__ RCim) {
  __shared__ float sYr[Cn][LSTR], sYi[Cn][LSTR];
  __shared__ float swt[LSTR], szero[LSTR];
  int n = blockIdx.x, tid = threadIdx.x;

  const float* gr = yre + (size_t)n * Cn * Tn;
  const float* gi = yim + (size_t)n * Cn * Tn;
  for (int idx = tid; idx < Cn * LSTR; idx += CORR_T) {
    int c = idx / LSTR, t = idx % LSTR;
    float vr = 0.f, vi = 0.f;
    if (t < Tn) { vr = gr[c * Tn + t]; vi = gi[c * Tn + t]; }
    sYr[c][t] = vr;
    sYi[c][t] = vi;
  }
  for (int idx = tid; idx < LSTR; idx += CORR_T) {
    swt[idx] = (idx < TFRM) ? invp[(size_t)n * Tn + (DELAYN + TAPS - 1) + idx]
                            : 0.f;   // zero-padded K tail kills padding terms
    szero[idx] = 0.f;
  }
  __syncthreads();

  int wave = tid >> 5, lane = tid & 31;
  int kh = lane >> 4, m = lane & 15;        // wave32: K-half, row/col-in-tile
  // upper-triangle tile enumeration {(jt,ct) : ct >= jt}, ct=4 is RHS column
  const unsigned char JT[14] = {0,0,0,0,0, 1,1,1,1, 2,2,2, 3,3};
  const unsigned char CT[14] = {0,1,2,3,4, 1,2,3,4, 2,3,4, 3,4};
  int jt = JT[wave], ct = CT[wave];

  // A operand: row j=(k,d): U[j,t]=Re(Y[d][t+7-k])*w[t], V=Im(...)*w
  int j = jt * 16 + m, d = j & 7, kk = j >> 3;
  const float* pUr = &sYr[d][TAPS - 1 - kk];
  const float* pUi = &sYi[d][TAPS - 1 - kk];
  // B operand: col l<64 -> Psi col (kb,e); 64..71 -> target Y[e][t+10]; pad->0
  int l = ct * 16 + m;
  const float *pBr, *pBi;
  if (l < KD) {
    int e = l & 7, kb = l >> 3;
    pBr = &sYr[e][TAPS - 1 - kb];
    pBi = &sYi[e][TAPS - 1 - kb];
  } else if (l < NCOLS) {
    int e = l - KD;
    pBr = &sYr[e][DELAYN + TAPS - 1];
    pBi = &sYi[e][DELAYN + TAPS - 1];
  } else {
    pBr = szero;
    pBi = szero;
  }

  v8f arr = {}, aii = {}, ari = {}, air = {};
  int kbase = 2 * kh;
  for (int t0 = 0; t0 < KPAD; t0 += 4) {
    int k0 = t0 + kbase;                      // this lane's two K indices
    float w0 = swt[k0], w1 = swt[k0 + 1];
    v2f au = {pUr[k0] * w0, pUr[k0 + 1] * w1};   // Re(conj side)*w
    v2f av = {pUi[k0] * w0, pUi[k0 + 1] * w1};   // Im(P)*w (sign handled below)
    v2f br = {pBr[k0], pBr[k0 + 1]};
    v2f bi = {pBi[k0], pBi[k0 + 1]};
    arr = wmma4(au, br, arr);   // U*Qr
    aii = wmma4(av, bi, aii);   // V*Qi
    ari = wmma4(au, bi, ari);   // U*Qi
    air = wmma4(av, br, air);   // V*Qr
  }
  // conj(P)w * Q : Re = U*Qr + V*Qi ; Im = U*Qi - V*Qr
  v8f Mr = arr + aii;
  v8f Mi = ari - air;

#pragma unroll
  for (int v = 0; v < 8; ++v) {
    int row = jt * 16 + v + 8 * kh;           // C/D layout
    if (l < KD) {
      size_t o = ((size_t)n * KD + row) * KD + l;
      Rre[o] = Mr[v];
      Rim[o] = Mi[v];
    } else if (l < NCOLS) {
      size_t o = ((size_t)n * KD + row) * Cn + (l - KD);
      RCre[o] = Mr[v];
      RCim[o] = Mi[v];
    }
  }
}

// ---------------- K_D: 64x64 complex Gauss-Jordan solve in LDS --------------
// Only upper-triangle tiles of R exist in memory; mirror the rest (Hermitian).
#define SOLVE_T 288
__global__ void __launch_bounds__(SOLVE_T) k_solve(
    const float* __restrict__ Rre, const float* __restrict__ Rim,
    const float* __restrict__ RCre, const float* __restrict__ RCim,
    float* __restrict__ Gre, float* __restrict__ Gim) {
  __shared__ float Ar[KD][NCOLS + 1], Ai[KD][NCOLS + 1];
  __shared__ float fr[KD], fi[KD];
  __shared__ float piv[2];
  int n = blockIdx.x, tid = threadIdx.x;

  for (int idx = tid; idx < KD * NCOLS; idx += SOLVE_T) {
    int i = idx / NCOLS, jj = idx % NCOLS;
    float vr, vi;
    if (jj < KD) {
      if ((i >> 4) <= (jj >> 4)) {            // stored tile
        size_t o = ((size_t)n * KD + i) * KD + jj;
        vr = Rre[o]; vi = Rim[o];
      } else {                                 // Hermitian mirror
        size_t o = ((size_t)n * KD + jj) * KD + i;
        vr = Rre[o]; vi = -Rim[o];
      }
    } else {
      size_t o = ((size_t)n * KD + i) * Cn + (jj - KD);
      vr = RCre[o]; vi = RCim[o];
    }
    Ar[i][jj] = vr;
    Ai[i][jj] = vi;
  }
  __syncthreads();
  if (tid == 0) {                    // diagonal loading: trace(R)*1e-10
    float trc = 0.f;
    for (int i = 0; i < KD; ++i) trc += Ar[i][i];
    float eps = trc * EPSF;
    for (int i = 0; i < KD; ++i) Ar[i][i] += eps;
  }
  __syncthreads();

  for (int k = 0; k < KD; ++k) {
    if (tid == 0) {
      float ar = Ar[k][k], ai = Ai[k][k];
      float den = ar * ar + ai * ai;
      piv[0] = ar / den;
      piv[1] = -ai / den;
    }
    __syncthreads();
    if (tid < KD) {
      float xr = Ar[tid][k], xi = Ai[tid][k];
      float pr = piv[0], pq = piv[1];
      float vr = xr * pr - xi * pq, vi = xr * pq + xi * pr;
      if (tid == k) { vr = 0.f; vi = 0.f; }
      fr[tid] = vr;
      fi[tid] = vi;
    }
    __syncthreads();
    {
      int jj = tid % NCOLS;
      int q = tid / NCOLS;           // 0..3, 16 rows each
      if (jj >= k) {                 // cols < k already eliminated
        float pr = Ar[k][jj], pq = Ai[k][jj];
        int i0 = q * 16;
#pragma unroll
        for (int u = 0; u < 16; ++u) {
          int i = i0 + u;
          if (i != k) {
            float fr_ = fr[i], fi_ = fi[i];
            Ar[i][jj] -= fr_ * pr - fi_ * pq;
            Ai[i][jj] -= fr_ * pq + fi_ * pr;
          }
        }
      }
    }
    __syncthreads();
  }

  if (tid < KD) {
    int i = tid;
    float dr = Ar[i][i], di = Ai[i][i];
    float den = dr * dr + di * di;
    float ivr = dr / den, ivi = -di / den;
#pragma unroll
    for (int e = 0; e < Cn; ++e) {
      float xr = Ar[i][KD + e], xi = Ai[i][KD + e];
      size_t o = ((size_t)n * KD + i) * Cn + e;
      Gre[o] = xr * ivr - xi * ivi;
      Gim[o] = xr * ivi + xi * ivr;
    }
  }
}

// ---------------- K_E: enhanced = mask * (Y - G^T Y_delayed)  (WMMA f32) ----
#define EV_T 256
__global__ void __launch_bounds__(EV_T) k_apply(
    const float* __restrict__ yre, const float* __restrict__ yim,
    const float* __restrict__ Gre, const float* __restrict__ Gim,
    const int* __restrict__ ilens,
    float* __restrict__ ere, float* __restrict__ eim) {
  __shared__ float sYr[Cn][LSTR], sYi[Cn][LSTR];
  __shared__ float sXr[KD][Cn], sXi[KD][Cn];
  int n = blockIdx.x, tid = threadIdx.x;

  const float* gr = yre + (size_t)n * Cn * Tn;
  const float* gi = yim + (size_t)n * Cn * Tn;
  for (int idx = tid; idx < Cn * LSTR; idx += EV_T) {
    int c = idx / LSTR, t = idx % LSTR;
    float vr = 0.f, vi = 0.f;
    if (t < Tn) { vr = gr[c * Tn + t]; vi = gi[c * Tn + t]; }
    sYr[c][t] = vr;
    sYi[c][t] = vi;
  }
  for (int idx = tid; idx < KD * Cn; idx += EV_T) {
    sXr[idx >> 3][idx & 7] = Gre[(size_t)n * KD * Cn + idx];
    sXi[idx >> 3][idx & 7] = Gim[(size_t)n * KD * Cn + idx];
  }
  __syncthreads();

  int len = ilens[n / Fn];
  int wave = tid >> 5, lane = tid & 31;
  int kh = lane >> 4, m = lane & 15;
  const int NT = (Tn + 15) / 16;               // 38 column tiles

  for (int ctile = wave; ctile < NT; ctile += EV_T / 32) {
    int t = ctile * 16 + m;
    v8f arr = {}, aii = {}, ari = {}, air = {};
    for (int kc = 0; kc < KD / 4; ++kc) {
      int k0 = kc * 4 + 2 * kh;
      v2f axr, axi, br, bi;
      if (m < Cn) {                            // A = G^T: rows 8..15 are pad
        axr = (v2f){sXr[k0][m], sXr[k0 + 1][m]};
        axi = (v2f){sXi[k0][m], sXi[k0 + 1][m]};
      } else {
        axr = (v2f){0.f, 0.f};
        axi = (v2f){0.f, 0.f};
      }
#pragma unroll
      for (int v = 0; v < 2; ++v) {
        int kj = k0 + v;                       // j = (p,d)
        int p = kj >> 3, dd = kj & 7;
        int src = t - DELAYN - p;              // Y_tilde: zero-pad head
        bool ok = (src >= 0) && (src < Tn);
        br[v] = ok ? sYr[dd][src] : 0.f;
        bi[v] = ok ? sYi[dd][src] : 0.f;
      }
      arr = wmma4(axr, br, arr);
      aii = wmma4(axi, bi, aii);
      ari = wmma4(axr, bi, ari);
      air = wmma4(axi, br, air);
    }
    v8f Rr = arr - aii;                        // (Xr+iXi)(Yr+iYi)
    v8f Ri = ari + air;
    if (kh == 0) {                             // D rows 0..7 = channels
#pragma unroll
      for (int v = 0; v < 8; ++v) {
        if (t < Tn) {
          bool valid = t < len;
          size_t o = ((size_t)n * Cn + v) * Tn + t;
          ere[o] = valid ? (sYr[v][t] - Rr[v]) : 0.f;
          eim[o] = valid ? (sYi[v][t] - Ri[v]) : 0.f;
        }
      }
    }
  }
}

// ---------------- K_F: (N,C,T) -> (B,T,C,F,2) interleaved output ------------
__global__ void k_out(const float* __restrict__ ere,
                      const float* __restrict__ eim,
                      float2* __restrict__ out) {
  __shared__ float tr[16][17], ti[16][17];
  int b = blockIdx.z, c = blockIdx.y;
  int tt = blockIdx.x / 17, ft = blockIdx.x % 17;
  int tx = threadIdx.x, ty = threadIdx.y;
  int t = tt * 16 + tx, f = ft * 16 + ty;      // read: t contiguous
  if (t < Tn && f < Fn) {
    int n = b * Fn + f;
    size_t s = ((size_t)n * Cn + c) * Tn + t;
    tr[ty][tx] = ere[s];
    ti[ty][tx] = eim[s];
  }
  __syncthreads();
  int t2 = tt * 16 + ty, f2 = ft * 16 + tx;    // write: f contiguous (float2)
  if (t2 < Tn && f2 < Fn) {
    size_t o = ((size_t)(b * Tn + t2) * Cn + c) * Fn + f2;
    out[o] = make_float2(tr[tx][ty], ti[tx][ty]);
  }
}

// ---------------- host-side launch ------------------------------------------
extern "C" void kernel_launch(void* const* d_in, const int* in_sizes, int n_in,
                              void* d_out, int out_size, void* d_ws,
                              size_t ws_size, hipStream_t stream) {
  (void)in_sizes; (void)n_in; (void)out_size; (void)ws_size;
  const float* dre  = (const float*)d_in[0];
  const float* dimg = (const float*)d_in[1];
  const int*   ilen = (const int*)d_in[2];

  float* ws = (float*)d_ws;
  const size_t SZY = (size_t)NTOT * Cn * Tn;   // 9,868,800 floats
  const size_t SZP = (size_t)NTOT * Tn;
  const size_t SZR = (size_t)NTOT * KD * KD;
  const size_t SZC = (size_t)NTOT * KD * Cn;
  float* Yre = ws;             float* Yim = Yre + SZY;
  float* Ere = Yim + SZY;      float* Eim = Ere + SZY;
  float* INVP = Eim + SZY;
  float* Rre = INVP + SZP;     float* Rim = Rre + SZR;
  float* RCre = Rim + SZR;     float* RCim = RCre + SZC;
  float* Gre = RCim + SZC;     float* Gim = Gre + SZC;

  dim3 tb(16, 16);
  dim3 tg(38 * 17, Cn, Bsz);
  k_pack<<<tg, tb, 0, stream>>>(dre, dimg, Yre, Yim);

  for (int it = 0; it < 2; ++it) {
    const float* sre = (it == 0) ? Yre : Ere;  // power from prev enhanced
    const float* sim = (it == 0) ? Yim : Eim;
    int nthr = NTOT * Tn;
    k_power<<<(nthr + 255) / 256, 256, 0, stream>>>(sre, sim, INVP);
    k_corr<<<NTOT, CORR_T, 0, stream>>>(Yre, Yim, INVP, Rre, Rim, RCre, RCim);
    k_solve<<<NTOT, SOLVE_T, 0, stream>>>(Rre, Rim, RCre, RCim, Gre, Gim);
    k_apply<<<NTOT, EV_T, 0, stream>>>(Yre, Yim, Gre, Gim, ilen, Ere, Eim);
  }
  k_out<<<tg, tb, 0, stream>>>(Ere, Eim, (float2*)d_out);
}